// WaveletTransform_64415919505634
// MI455X (gfx1250) — compile-verified
//
#include <hip/hip_runtime.h>

// db4 wavelet roundtrip (pywt symmetric mode) on (8,32,512,512) fp32.
// All filtering mapped to V_WMMA_F32_16X16X4_F32 tile GEMMs (fp32 matrix pipe).
// Filter taps live in LDS; banded-matrix fragments are branchless selects.

typedef __attribute__((ext_vector_type(2))) float v2f;
typedef __attribute__((ext_vector_type(8))) float v8f;

#define NN 512          // H == W
#define MM 259          // dwt coeff length: (512+7)/2
#define WS_STRIDE 260   // padded row stride for subband planes
#define NSLICE 256      // B*C slices
#define TILES_C 17      // ceil(259/16)

// Concatenated taps: [0..7]=GA_LO (dec_lo reversed), [8..15]=GA_HI,
//                    [16..23]=DEC_LO, [24..31]=DEC_HI.
__device__ const float GTAPS[32] = {
  // GA_LO = dec_lo reversed (analysis lowpass correlation taps)
   0.23037781330885523f,  0.7148465705525415f,   0.6308807679295904f,
  -0.02798376941698385f, -0.18703481171888114f,  0.030841381835986965f,
   0.032883011666982945f, -0.010597401784997278f,
  // GA_HI = dec_hi reversed
  -0.010597401784997278f, -0.032883011666982945f, 0.030841381835986965f,
   0.18703481171888114f,  -0.02798376941698385f, -0.6308807679295904f,
   0.7148465705525415f,   -0.23037781330885523f,
  // DEC_LO (synthesis correlation taps)
  -0.010597401784997278f,  0.032883011666982945f,  0.030841381835986965f,
  -0.18703481171888114f,  -0.02798376941698385f,   0.6308807679295904f,
   0.7148465705525415f,    0.23037781330885523f,
  // DEC_HI
  -0.23037781330885523f,   0.7148465705525415f,   -0.6308807679295904f,
  -0.02798376941698385f,   0.18703481171888114f,   0.030841381835986965f,
  -0.032883011666982945f, -0.010597401784997278f };

#define T_GALO  0
#define T_GAHI  8
#define T_DECLO 16
#define T_DECHI 24

__device__ __forceinline__ int symmap(int i) {
  if (i < 0)   i = -1 - i;          // symmetric (half-sample) reflection
  if (i >= NN) i = 2 * NN - 1 - i;
  return i;
}
__device__ __forceinline__ int iclampM(int i) { return i < (MM - 1) ? i : (MM - 1); }

// Branchless banded lookup: g[j] if 0<=j<8 (and extra condition) else 0.
// FT points at the 32-entry LDS tap table; index is always in-bounds.
__device__ __forceinline__ float tapsel(const float* FT, int base, int j, bool ok) {
  float v = FT[base + (j & 7)];
  return (((unsigned)j < 8u) && ok) ? v : 0.0f;
}

__device__ __forceinline__ v8f wmma4(v2f a, v2f b, v8f c) {
  return __builtin_amdgcn_wmma_f32_16x16x4_f32(false, a, false, b,
                                               (short)0, c, false, false);
}

// ---------------- Analysis: x -> LL/LH/HL/HH tiles -------------------------
__global__ __launch_bounds__(32)
void wavelet_analysis(const float* __restrict__ x, float* __restrict__ ws) {
  const int lane  = threadIdx.x;
  const int tile  = blockIdx.x;          // 0..288
  const int slice = blockIdx.y;          // 0..255
  const int ty = tile / TILES_C, tx = tile % TILES_C;
  const int hc0 = ty * 16, wc0 = tx * 16;
  const int rbase = 2 * hc0 - 5, cbase = 2 * wc0 - 5;   // out[o] uses x[2o-5+j]

  __shared__ float FT[32];         // tap table
  __shared__ float X[40 * 41];     // 40x40 input tile (stride 41, bank-safe)
  __shared__ float Tlo[40 * 16];   // W-lowpass intermediate (rows x 16 wc)
  __shared__ float Thi[40 * 16];   // W-highpass intermediate

  FT[lane] = GTAPS[lane];
  const float* xs = x + (size_t)slice * (NN * NN);
  for (int p = lane; p < 1600; p += 32) {
    int r = p / 40, c = p % 40;
    X[r * 41 + c] = xs[symmap(rbase + r) * NN + symmap(cbase + c)];
  }
  __syncthreads();

  const int half = lane >> 4;      // 0: K=0,1   1: K=2,3 of each 4-chunk
  const int l16  = lane & 15;

  // Banded filter fragments G[k][n] = g[k - 2n]; identical value serves as
  // stage-1 B operand and stage-2 A operand. Computed once per block.
  v2f fLO[10], fHI[10];
#pragma unroll
  for (int kc = 0; kc < 10; ++kc) {
    const int j0 = kc * 4 + 2 * half - 2 * l16;
    fLO[kc].x = tapsel(FT, T_GALO, j0,     true);
    fLO[kc].y = tapsel(FT, T_GALO, j0 + 1, true);
    fHI[kc].x = tapsel(FT, T_GAHI, j0,     true);
    fHI[kc].y = tapsel(FT, T_GAHI, j0 + 1, true);
  }

  // Stage 1: W-direction analysis. D(16 rows x 16 wc) = X(16x40) * G(40x16).
  const int rowblocks[3] = {0, 16, 24};   // covers rows 0..39 (overlap benign)
#pragma unroll
  for (int rbI = 0; rbI < 3; ++rbI) {
    const int rb = rowblocks[rbI];
    v8f dlo = {}; v8f dhi = {};
#pragma unroll
    for (int kc = 0; kc < 10; ++kc) {
      const int k0 = kc * 4 + 2 * half;
      v2f a;                        // A[M=row][K=sample] from LDS
      a.x = X[(rb + l16) * 41 + k0];
      a.y = X[(rb + l16) * 41 + k0 + 1];
      dlo = wmma4(a, fLO[kc], dlo);
      dhi = wmma4(a, fHI[kc], dhi);
    }
#pragma unroll
    for (int r = 0; r < 8; ++r) {   // D row m = r + 8*half, col n = l16
      const int rt = rb + r + 8 * half;
      Tlo[rt * 16 + l16] = dlo[r];
      Thi[rt * 16 + l16] = dhi[r];
    }
  }
  __syncthreads();

  // Stage 2: H-direction analysis. D(16 hc x 16 wc) = G(16x40) * T(40x16).
  v8f dLL = {}, dLH = {}, dHL = {}, dHH = {};
#pragma unroll
  for (int kc = 0; kc < 10; ++kc) {
    const int k0 = kc * 4 + 2 * half;
    v2f bLO, bHI;                   // B[K=row][N=wc] data
    bLO.x = Tlo[(k0) * 16 + l16];   bLO.y = Tlo[(k0 + 1) * 16 + l16];
    bHI.x = Thi[(k0) * 16 + l16];   bHI.y = Thi[(k0 + 1) * 16 + l16];
    dLL = wmma4(fLO[kc], bLO, dLL); dLH = wmma4(fHI[kc], bLO, dLH);
    dHL = wmma4(fLO[kc], bHI, dHL); dHH = wmma4(fHI[kc], bHI, dHH);
  }

  const size_t ps = (size_t)NSLICE * MM * WS_STRIDE;
  float* LL = ws;            float* LH = ws + ps;
  float* HL = ws + 2 * ps;   float* HH = ws + 3 * ps;
  const size_t sb = (size_t)slice * MM * WS_STRIDE;
  const int wc = wc0 + l16;
  if (wc < MM) {
#pragma unroll
    for (int r = 0; r < 8; ++r) {
      const int hc = hc0 + r + 8 * half;
      if (hc < MM) {
        const size_t o = sb + (size_t)hc * WS_STRIDE + wc;
        LL[o] = dLL[r]; LH[o] = dLH[r]; HL[o] = dHL[r]; HH[o] = dHH[r];
      }
    }
  }
}

// ---------------- Synthesis: subbands -> rec -------------------------------
__global__ __launch_bounds__(32)
void wavelet_synthesis(const float* __restrict__ ws, float* __restrict__ out) {
  const int lane  = threadIdx.x;
  const int tile  = blockIdx.x;          // 0..1023
  const int slice = blockIdx.y;
  const int ty = tile >> 5, tx = tile & 31;
  const int h0 = ty * 16, w0 = tx * 16;
  const int i0h = h0 >> 1, i0w = w0 >> 1;   // first coeff feeding this tile
  const int half = lane >> 4, l16 = lane & 15;

  __shared__ float FT[32];
  __shared__ float LoW[16 * 17];   // idwt-H lowpass  (h x wc), stride 17
  __shared__ float HiW[16 * 17];   // idwt-H highpass

  FT[lane] = GTAPS[lane];
  __syncthreads();

  const size_t ps = (size_t)NSLICE * MM * WS_STRIDE;
  const float* LL = ws;            const float* LH = ws + ps;
  const float* HL = ws + 2 * ps;   const float* HH = ws + 3 * ps;
  const size_t sb = (size_t)slice * MM * WS_STRIDE;

  // Stage 1 (along H): LoW = S_lo(16x12)*LLblk + S_hi*LHblk ; HiW likewise.
  // S[n][b] = dec[2b - n + 1] if tap valid and i0+b < MM.
  v8f dlo = {}, dhi = {};
#pragma unroll
  for (int kc = 0; kc < 3; ++kc) {
    const int k0 = kc * 4 + 2 * half;
    const int t0 = 2 * k0 - l16 + 1;          // tap index for b=k0, n=l16
    v2f aL, aH;                               // A[M=h][K=b]
    aL.x = tapsel(FT, T_DECLO, t0,     (i0h + k0)     < MM);
    aL.y = tapsel(FT, T_DECLO, t0 + 2, (i0h + k0 + 1) < MM);
    aH.x = tapsel(FT, T_DECHI, t0,     (i0h + k0)     < MM);
    aH.y = tapsel(FT, T_DECHI, t0 + 2, (i0h + k0 + 1) < MM);

    // B[K=b][N=wc] coeff data: clamp address in-plane, select to zero.
    const int r0 = i0h + k0, r1 = r0 + 1;
    const int cc = i0w + l16;
    const bool cok = cc < MM;
    const bool v0 = (r0 < MM) && cok, v1 = (r1 < MM) && cok;
    const size_t o0 = sb + (size_t)iclampM(r0) * WS_STRIDE + iclampM(cc);
    const size_t o1 = sb + (size_t)iclampM(r1) * WS_STRIDE + iclampM(cc);
    v2f bLL, bLH, bHL, bHH;
    bLL.x = v0 ? LL[o0] : 0.f;  bLL.y = v1 ? LL[o1] : 0.f;
    bLH.x = v0 ? LH[o0] : 0.f;  bLH.y = v1 ? LH[o1] : 0.f;
    bHL.x = v0 ? HL[o0] : 0.f;  bHL.y = v1 ? HL[o1] : 0.f;
    bHH.x = v0 ? HH[o0] : 0.f;  bHH.y = v1 ? HH[o1] : 0.f;
    dlo = wmma4(aL, bLL, dlo);  dlo = wmma4(aH, bLH, dlo);
    dhi = wmma4(aL, bHL, dhi);  dhi = wmma4(aH, bHH, dhi);
  }
#pragma unroll
  for (int r = 0; r < 8; ++r) {     // reshape D -> A operand via LDS
    const int a = r + 8 * half;
    LoW[a * 17 + l16] = dlo[r];
    HiW[a * 17 + l16] = dhi[r];
  }
  __syncthreads();

  // Stage 2 (along W): rec(16x16) = LoW(16x12)*S_lo + HiW(16x12)*S_hi.
  v8f drec = {};
#pragma unroll
  for (int kc = 0; kc < 3; ++kc) {
    const int k0 = kc * 4 + 2 * half;
    const int t0 = 2 * k0 - l16 + 1;
    v2f aLo, aHi, bL, bH;
    aLo.x = LoW[l16 * 17 + k0];  aLo.y = LoW[l16 * 17 + k0 + 1];
    aHi.x = HiW[l16 * 17 + k0];  aHi.y = HiW[l16 * 17 + k0 + 1];
    bL.x = tapsel(FT, T_DECLO, t0,     (i0w + k0)     < MM);
    bL.y = tapsel(FT, T_DECLO, t0 + 2, (i0w + k0 + 1) < MM);
    bH.x = tapsel(FT, T_DECHI, t0,     (i0w + k0)     < MM);
    bH.y = tapsel(FT, T_DECHI, t0 + 2, (i0w + k0 + 1) < MM);
    drec = wmma4(aLo, bL, drec);
    drec = wmma4(aHi, bH, drec);
  }

  float* o = out + (size_t)slice * NN * NN;
#pragma unroll
  for (int r = 0; r < 8; ++r) {
    const int h = h0 + r + 8 * half;
    o[(size_t)h * NN + w0 + l16] = drec[r];
  }
}

extern "C" void kernel_launch(void* const* d_in, const int* in_sizes, int n_in,
                              void* d_out, int out_size, void* d_ws, size_t ws_size,
                              hipStream_t stream) {
  (void)in_sizes; (void)n_in; (void)out_size; (void)ws_size;
  const float* x = (const float*)d_in[0];        // (8,32,512,512) fp32
  float* out = (float*)d_out;                    // (8,32,512,512) fp32
  float* ws  = (float*)d_ws;                     // needs 4*256*259*260*4 B ~ 276 MB

  dim3 blk(32, 1, 1);                            // one wave32 per tile
  dim3 gA(TILES_C * TILES_C, NSLICE, 1);         // 17x17 coeff tiles per slice
  dim3 gS(32 * 32, NSLICE, 1);                   // 32x32 output tiles per slice
  wavelet_analysis<<<gA, blk, 0, stream>>>(x, ws);
  wavelet_synthesis<<<gS, blk, 0, stream>>>(ws, out);
}